// TaylorExp_23983097380979
// MI455X (gfx1250) — compile-verified
//
#include <hip/hip_runtime.h>

typedef __attribute__((ext_vector_type(2))) float v2f;
typedef __attribute__((ext_vector_type(8))) float v8f;

// Each wave: 8 sequence positions, one V_WMMA_F32_16X16X4_F32 outer product each.
// Output tile layout of WMMA (VGPR r -> rows M=r / M=r+8, N striped over lanes)
// maps 1:1 onto the flattened d*d output row, so stores are plain coalesced
// b32 stores covering contiguous 64B rows.
#define POS_PER_WAVE 8
#define WAVES_PER_BLOCK 8

__global__ __launch_bounds__(256) void taylor_outer_wmma(
    const float* __restrict__ x, float* __restrict__ out, int npos)
{
    const int lane = threadIdx.x & 31;
    const int wave = threadIdx.x >> 5;
    const int c16  = lane & 15;
    const bool hi  = lane >= 16;   // lanes 16..31

    const int base = (blockIdx.x * WAVES_PER_BLOCK + wave) * POS_PER_WAVE;
    if (base >= npos) return;      // wave-uniform guard (EXEC stays all-1s)

    #pragma unroll
    for (int i = 0; i < POS_PER_WAVE; ++i) {
        const int p = base + i;

        // Load x[p][0..15]; lanes 16-31 duplicate the same 64B line (coalesced).
        const float xv = x[(size_t)p * 16 + c16];

        // A (16x4): column K=0 = x * 0.25 (folds the 1/sqrt(16) scale), K=1..3 = 0.
        // Lanes 16-31 hold K=2/K=3 -> must be zero. v_cndmask, no EXEC change.
        const float av = hi ? 0.0f : xv * 0.25f;
        // B (4x16): row K=0 = x, K=1..3 = 0 (lanes 16-31 hold K=2/K=3).
        const float bv = hi ? 0.0f : xv;

        v2f A;  A[0]  = av; A[1]  = 0.0f;
        v2f Bm; Bm[0] = bv; Bm[1] = 0.0f;
        v8f C = {};

        // D[M][N] = (x[M]/4) * x[N]  -- exact f32 outer product
        v8f D = __builtin_amdgcn_wmma_f32_16x16x4_f32(
            /*neg_a=*/false, A, /*neg_b=*/false, Bm,
            /*c_mod=*/(short)0, C, /*reuse_a=*/false, /*reuse_b=*/false);

        // VGPR r holds out[r*16 + N] (lanes 0-15) and out[(r+8)*16 + N] (lanes 16-31).
        float* op = out + (size_t)p * 256 + c16 + (hi ? 128 : 0);
        #pragma unroll
        for (int r = 0; r < 8; ++r)
            __builtin_nontemporal_store(D[r], op + r * 16);  // NT: 256MiB > L2, never re-read
    }
}

extern "C" void kernel_launch(void* const* d_in, const int* in_sizes, int n_in,
                              void* d_out, int out_size, void* d_ws, size_t ws_size,
                              hipStream_t stream)
{
    const float* x = (const float*)d_in[0];
    float* out = (float*)d_out;

    const int n    = in_sizes[0];   // B*H*S*D = 4,194,304
    const int npos = n / 16;        // 262,144 outer products
    const int pos_per_block = POS_PER_WAVE * WAVES_PER_BLOCK;  // 64
    const int blocks = (npos + pos_per_block - 1) / pos_per_block;

    taylor_outer_wmma<<<blocks, 256, 0, stream>>>(x, out, npos);
}